// TorchModelGeometry_40836549050715
// MI455X (gfx1250) — compile-verified
//
#include <hip/hip_runtime.h>
#include <math.h>

typedef __attribute__((ext_vector_type(2))) float v2f;
typedef __attribute__((ext_vector_type(8))) float v8f;

#define NVIS 16384
#define NTAC 2048
#define NMDL 8192
#define VIS_TILES (NVIS / 16)
#define TAC_TILES (NTAC / 16)
#define MDL_TILES (NMDL / 16)

// ---------------------------------------------------------------------------
// Kernel 1: build rotation from state, transform+pad points, precompute norms.
// Also zeroes the output accumulator (runs before the chamfer kernel on the
// same stream, so ordering is guaranteed).
// ---------------------------------------------------------------------------
__global__ void transform_pad_kernel(const float* __restrict__ vis,
                                     const float* __restrict__ tac,
                                     const float* __restrict__ mdl,
                                     const float* __restrict__ scale_p,
                                     const float* __restrict__ state,
                                     float* __restrict__ vis_pad,
                                     float* __restrict__ tac_pad,
                                     float* __restrict__ mdl_pad,
                                     float* __restrict__ a_vis,
                                     float* __restrict__ a_tac,
                                     float* __restrict__ b_mdl,
                                     float* __restrict__ out) {
    const int gid = blockIdx.x * blockDim.x + threadIdx.x;
    if (gid == 0) out[0] = 0.0f;
    const int total = NVIS + NTAC + NMDL;
    if (gid >= total) return;

    if (gid >= NVIS + NTAC) {
        // model points: pad only, compute b = ||p||^2
        const int j = gid - (NVIS + NTAC);
        const float x = mdl[j * 3 + 0];
        const float y = mdl[j * 3 + 1];
        const float z = mdl[j * 3 + 2];
        mdl_pad[j * 4 + 0] = x;
        mdl_pad[j * 4 + 1] = y;
        mdl_pad[j * 4 + 2] = z;
        mdl_pad[j * 4 + 3] = 0.0f;
        b_mdl[j] = x * x + y * y + z * z;
        return;
    }

    // Rotation R = Rz @ Ry @ Rx (each thread recomputes; cheap)
    const float tx = state[0], ty = state[1], tz = state[2];
    const float sx = sinf(state[3]), cx = cosf(state[3]);
    const float sy = sinf(state[4]), cy = cosf(state[4]);
    const float sz = sinf(state[5]), cz = cosf(state[5]);
    // Columns of R (vc = d @ R => vc_k = d . R[:,k])
    const float r00 = cz * cy;
    const float r10 = sz * cy;
    const float r20 = -sy;
    const float r01 = cz * sy * sx - sz * cx;
    const float r11 = cz * cx + sz * sy * sx;
    const float r21 = cy * sx;
    const float r02 = sz * sx + cz * sy * cx;
    const float r12 = sz * sy * cx - cz * sx;
    const float r22 = cy * cx;
    const float inv_s = 1.0f / scale_p[0];

    const float* src;
    float* dst;
    float* anorm;
    int i;
    if (gid < NVIS) {
        src = vis; dst = vis_pad; anorm = a_vis; i = gid;
    } else {
        src = tac; dst = tac_pad; anorm = a_tac; i = gid - NVIS;
    }
    const float dx = src[i * 3 + 0] - tx;
    const float dy = src[i * 3 + 1] - ty;
    const float dz = src[i * 3 + 2] - tz;
    const float px = (dx * r00 + dy * r10 + dz * r20) * inv_s;
    const float py = (dx * r01 + dy * r11 + dz * r21) * inv_s;
    const float pz = (dx * r02 + dy * r12 + dz * r22) * inv_s;
    dst[i * 4 + 0] = px;
    dst[i * 4 + 1] = py;
    dst[i * 4 + 2] = pz;
    dst[i * 4 + 3] = 0.0f;
    anorm[i] = px * px + py * py + pz * pz;
}

// ---------------------------------------------------------------------------
// Kernel 2: chamfer distance via V_WMMA_F32_16X16X4_F32.
// One wave32 per 16-row pred tile; loops over 512 model column tiles.
// A-frag : lane l holds pred row (l&15), K-pair 2*(l>>4)   -> float2 load
// B-frag : lane l holds model col (l&15), K-pair 2*(l>>4)  -> float2 load
// C/D    : lane l, vgpr v holds M = v + 8*(l>>4), N = l&15
// dist(m,n) = a[m] + b[n] - 2*c  ->  track min_n(b[n] - 2c), add a[m] at end.
// ---------------------------------------------------------------------------
__global__ void __launch_bounds__(256)
chamfer_wmma_kernel(const float* __restrict__ vis_pad,
                    const float* __restrict__ tac_pad,
                    const float* __restrict__ a_vis,
                    const float* __restrict__ a_tac,
                    const float* __restrict__ mdl_pad,
                    const float* __restrict__ b_mdl,
                    float* __restrict__ out) {
    const int wave = (blockIdx.x * blockDim.x + threadIdx.x) >> 5;
    const int lane = threadIdx.x & 31;
    const int half = lane >> 4;   // 0: K=0..1 / rows M=0..7 ; 1: K=2..3 / rows M=8..15
    const int n0   = lane & 15;

    const float* pred;
    const float* anorm;
    int tile;
    float w;
    if (wave < VIS_TILES) {
        pred = vis_pad; anorm = a_vis; tile = wave;             w = 1.0f / (float)NVIS;
    } else {
        pred = tac_pad; anorm = a_tac; tile = wave - VIS_TILES; w = 0.1f / (float)NTAC;
    }
    const int rowBase = tile * 16;

    // A fragment: pred row (lane&15), K-pair (half)
    const v2f afrag = *(const v2f*)(pred + (size_t)(rowBase + n0) * 4 + 2 * half);

    float mn[8];
#pragma unroll
    for (int v = 0; v < 8; ++v) mn[v] = 3.402823466e+38f;

    for (int jt = 0; jt < MDL_TILES; ++jt) {
        const int col = jt * 16 + n0;
        const v2f bfrag = *(const v2f*)(mdl_pad + (size_t)col * 4 + 2 * half);
        const float bb = b_mdl[col];

        v8f c = {};
        c = __builtin_amdgcn_wmma_f32_16x16x4_f32(
                /*neg_a=*/false, afrag,
                /*neg_b=*/false, bfrag,
                /*c_mod=*/(short)0, c,
                /*reuse_a=*/false, /*reuse_b=*/false);

#pragma unroll
        for (int v = 0; v < 8; ++v)
            mn[v] = fminf(mn[v], __builtin_fmaf(-2.0f, c[v], bb));
    }

    // Min across the 16 lanes of each half (butterfly stays within halves).
#pragma unroll
    for (int m = 1; m < 16; m <<= 1) {
#pragma unroll
        for (int v = 0; v < 8; ++v)
            mn[v] = fminf(mn[v], __shfl_xor(mn[v], m, 32));
    }

    // Lane 0 of each half: sum (min + a[m]) over its 8 rows.
    float partial = 0.0f;
    if (n0 == 0) {
#pragma unroll
        for (int v = 0; v < 8; ++v)
            partial += mn[v] + anorm[rowBase + half * 8 + v];
    }
    partial += __shfl_xor(partial, 16, 32);   // combine the two halves into lane 0
    if (lane == 0) atomicAdd(out, w * partial);
}

// ---------------------------------------------------------------------------
extern "C" void kernel_launch(void* const* d_in, const int* in_sizes, int n_in,
                              void* d_out, int out_size, void* d_ws, size_t ws_size,
                              hipStream_t stream) {
    const float* vis   = (const float*)d_in[0];  // (16384,3)
    const float* tac   = (const float*)d_in[1];  // (2048,3)
    const float* mdl   = (const float*)d_in[2];  // (8192,3)
    const float* scale = (const float*)d_in[3];  // ()
    const float* state = (const float*)d_in[4];  // (6,)
    float* out = (float*)d_out;

    float* ws       = (float*)d_ws;
    float* vis_pad  = ws;                       // NVIS*4
    float* tac_pad  = vis_pad + NVIS * 4;       // NTAC*4
    float* mdl_pad  = tac_pad + NTAC * 4;       // NMDL*4
    float* a_vis    = mdl_pad + NMDL * 4;       // NVIS
    float* a_tac    = a_vis + NVIS;             // NTAC
    float* b_mdl    = a_tac + NTAC;             // NMDL

    {
        const int total = NVIS + NTAC + NMDL;
        const int blk = 256;
        const int grd = (total + blk - 1) / blk;
        transform_pad_kernel<<<grd, blk, 0, stream>>>(
            vis, tac, mdl, scale, state,
            vis_pad, tac_pad, mdl_pad, a_vis, a_tac, b_mdl, out);
    }
    {
        const int nWaves = VIS_TILES + TAC_TILES;     // 1152
        const int blk = 256;                          // 8 waves per block
        const int grd = nWaves / (blk / 32);          // 144
        chamfer_wmma_kernel<<<grd, blk, 0, stream>>>(
            vis_pad, tac_pad, a_vis, a_tac, mdl_pad, b_mdl, out);
    }
}